// AttentionFreeTransformerEncoder_41085657153701
// MI455X (gfx1250) — compile-verified
//
#include <hip/hip_runtime.h>
#include <math.h>

// ---------------------------------------------------------------------------
// AFT-full transformer encoder block for MI455X (gfx1250, wave32).
// GEMMs: V_WMMA_F32_16X16X4_F32 (exact f32), 128x128 block tiles, 32x64 wave
// tiles (8 accumulators), async global->LDS (ASYNCcnt) double-buffered
// staging. Right operands are pre-transposed to [N][K] so every WMMA fragment
// is a single aligned ds_load_b64 (no operand-packing movs).
// B=8, T=2048, D=512, H=2048.
// ---------------------------------------------------------------------------

typedef __attribute__((ext_vector_type(2))) float v2f;
typedef __attribute__((ext_vector_type(8))) float v8f;

namespace {

constexpr int Bc = 8, Tc = 2048, Dc = 512, Hc = 2048;
constexpr float LN_EPS = 1e-5f;

enum { EPI_NONE = 0, EPI_RES = 1, EPI_GELU = 2, EPI_GELU_RES = 3 };

__device__ __forceinline__ float gelu_exact(float x) {
  return 0.5f * x * (1.0f + erff(x * 0.70710678118654752440f));
}

__device__ __forceinline__ float wave_sum(float v) {
#pragma unroll
  for (int off = 16; off > 0; off >>= 1) v += __shfl_xor(v, off, 32);
  return v;
}

__device__ __forceinline__ float wave_max(float v) {
#pragma unroll
  for (int off = 16; off > 0; off >>= 1) v = fmaxf(v, __shfl_xor(v, off, 32));
  return v;
}

// Async DMA: copy 16 bytes from (base + voff) in global memory into LDS at
// byte offset lds_off. Tracked by ASYNCcnt, completes in issue order.
__device__ __forceinline__ void async_b128(unsigned lds_off, const float* base,
                                           unsigned voff) {
  asm volatile("global_load_async_to_lds_b128 %0, %1, %2"
               :: "v"(lds_off), "v"(voff), "s"(base)
               : "memory");
}

__device__ __forceinline__ unsigned lds_addr_of(const float* p) {
  return (unsigned)(unsigned long long)p;  // low 32 bits of LDS aperture addr
}

__device__ __forceinline__ v2f ld2(const float* p) {
  return *(const v2f*)p;  // 8B-aligned: even float offset
}

// ------------------------- LayerNorm (D = 512) -----------------------------
__global__ void ln_kernel(const float* __restrict__ x, const float* __restrict__ g,
                          const float* __restrict__ b, float* __restrict__ o) {
  const int row  = blockIdx.x * 8 + (threadIdx.x >> 5);
  const int lane = threadIdx.x & 31;
  const float4* xr = (const float4*)(x + (size_t)row * Dc);
  float4 v[4];
  float s = 0.0f, ss = 0.0f;
#pragma unroll
  for (int i = 0; i < 4; ++i) {
    v[i] = xr[i * 32 + lane];
    s  += v[i].x + v[i].y + v[i].z + v[i].w;
    ss += v[i].x * v[i].x + v[i].y * v[i].y + v[i].z * v[i].z + v[i].w * v[i].w;
  }
  s = wave_sum(s);
  ss = wave_sum(ss);
  const float mu   = s * (1.0f / Dc);
  const float var  = ss * (1.0f / Dc) - mu * mu;
  const float rstd = rsqrtf(var + LN_EPS);
  float4* orow = (float4*)(o + (size_t)row * Dc);
  const float4* g4 = (const float4*)g;
  const float4* b4 = (const float4*)b;
#pragma unroll
  for (int i = 0; i < 4; ++i) {
    const float4 gg = g4[i * 32 + lane];
    const float4 bb = b4[i * 32 + lane];
    const float4 vv = v[i];
    float4 r;
    r.x = (vv.x - mu) * rstd * gg.x + bb.x;
    r.y = (vv.y - mu) * rstd * gg.y + bb.y;
    r.z = (vv.z - mu) * rstd * gg.z + bb.z;
    r.w = (vv.w - mu) * rstd * gg.w + bb.w;
    orow[i * 32 + lane] = r;
  }
}

// ------------------------- Tiled transpose ---------------------------------
// out[Cc][R] = in[R][Cc]^T, batched. Block 256 = 32x8, 32x32 tiles.
__global__ void transpose_kernel(const float* __restrict__ in, float* __restrict__ out,
                                 int R, int Cc, long long inB, long long outB) {
  __shared__ float t[32][33];
  const float* ib = in + (long long)blockIdx.z * inB;
  float* ob = out + (long long)blockIdx.z * outB;
  const int c0 = blockIdx.x * 32, r0 = blockIdx.y * 32;
  const int tx = threadIdx.x & 31, ty = threadIdx.x >> 5;
#pragma unroll
  for (int i = 0; i < 32; i += 8)
    t[ty + i][tx] = ib[(size_t)(r0 + ty + i) * Cc + c0 + tx];
  __syncthreads();
#pragma unroll
  for (int i = 0; i < 32; i += 8)
    ob[(size_t)(c0 + ty + i) * R + r0 + tx] = t[tx][ty + i];
}

// ---------------- WMMA f32 GEMM: C = A @ Wt^T (+bias, epilogue) ------------
// A:  [M][K] row-major.  Wt: [N][K] row-major (pre-transposed weights).
// 256 threads = 8 wave32. Block tile 128(M) x 128(N); wave tile 32x64
// (2x4 accumulators). K staged 16-wide, double-buffered async-to-LDS DMA.
constexpr int KC  = 16;
constexpr int LDA = 20;               // [128][20] padded tiles (bank-safe)
constexpr int TILE = 128 * LDA;

template <int EPI>
__global__ __launch_bounds__(256)
void gemm_wmma_f32(const float* __restrict__ A, const float* __restrict__ Wt,
                   const float* __restrict__ bias, const float* __restrict__ res,
                   float* __restrict__ C, int M, int N, int Kd,
                   long long aBatch, long long wBatch, long long cBatch) {
  __shared__ float As[2][TILE];
  __shared__ float Bs[2][TILE];

  const int tid  = threadIdx.x;
  const int wave = tid >> 5;
  const int lane = tid & 31;
  const int half = lane >> 4;   // 0: lanes 0-15, 1: lanes 16-31
  const int ln   = lane & 15;

  const int m_blk = blockIdx.y * 128;
  const int n_blk = blockIdx.x * 128;
  const long long batch = blockIdx.z;
  A  += batch * aBatch;
  Wt += batch * wBatch;
  C  += batch * cBatch;
  const float* resb = res ? res + batch * cBatch : nullptr;

  const int m_off = (wave >> 1) * 32;  // 0/32/64/96
  const int n_off = (wave & 1) * 64;   // 0/64

  v8f acc[2][4] = {};

  // --- staging: each thread DMAs 4x16B per K-chunk --------------------------
  const int srow = tid >> 2;          // 0..63 (and +64 for second half)
  const int scol = (tid & 3) * 4;     // 0,4,8,12

  unsigned aoff0 = ((unsigned)(m_blk + srow) * (unsigned)Kd + (unsigned)scol) * 4u;
  unsigned aoff1 = aoff0 + (unsigned)(64 * Kd) * 4u;
  unsigned boff0 = ((unsigned)(n_blk + srow) * (unsigned)Kd + (unsigned)scol) * 4u;
  unsigned boff1 = boff0 + (unsigned)(64 * Kd) * 4u;

  unsigned lA0[2], lA1[2], lB0[2], lB1[2];
#pragma unroll
  for (int b = 0; b < 2; ++b) {
    lA0[b] = lds_addr_of(&As[b][srow * LDA + scol]);
    lA1[b] = lds_addr_of(&As[b][(srow + 64) * LDA + scol]);
    lB0[b] = lds_addr_of(&Bs[b][srow * LDA + scol]);
    lB1[b] = lds_addr_of(&Bs[b][(srow + 64) * LDA + scol]);
  }

  auto issue = [&](int buf) {
    async_b128(lA0[buf], A,  aoff0);
    async_b128(lA1[buf], A,  aoff1);
    async_b128(lB0[buf], Wt, boff0);
    async_b128(lB1[buf], Wt, boff1);
    aoff0 += KC * 4u; aoff1 += KC * 4u;
    boff0 += KC * 4u; boff1 += KC * 4u;
  };

  auto compute = [&](const float* Ab, const float* Bb) {
#pragma unroll
    for (int ks = 0; ks < 4; ++ks) {
      // 16x16x4 fragments: lanes 0-15 hold K={4ks,4ks+1},
      // lanes 16-31 hold K={4ks+2,4ks+3}; both operands are [row][k] in LDS.
      const int k = ks * 4 + 2 * half;
      v2f a[2], bfr[4];
#pragma unroll
      for (int mi = 0; mi < 2; ++mi)
        a[mi] = ld2(&Ab[(m_off + 16 * mi + ln) * LDA + k]);
#pragma unroll
      for (int nj = 0; nj < 4; ++nj)
        bfr[nj] = ld2(&Bb[(n_off + 16 * nj + ln) * LDA + k]);
#pragma unroll
      for (int mi = 0; mi < 2; ++mi)
#pragma unroll
        for (int nj = 0; nj < 4; ++nj)
          acc[mi][nj] = __builtin_amdgcn_wmma_f32_16x16x4_f32(
              false, a[mi], false, bfr[nj], (short)0, acc[mi][nj], false, false);
    }
  };

  const int nch = Kd / KC;
  issue(0);
  int cur = 0;
  for (int c = 0; c < nch - 1; ++c) {
    issue(cur ^ 1);
    // Leave only the 4 just-issued DMAs in flight -> chunk `cur` is in LDS.
    asm volatile("s_wait_asynccnt 0x4" ::: "memory");
    __syncthreads();
    compute(As[cur], Bs[cur]);
    __syncthreads();
    cur ^= 1;
  }
  asm volatile("s_wait_asynccnt 0x0" ::: "memory");
  __syncthreads();
  compute(As[cur], Bs[cur]);

  // Epilogue. C/D layout: VGPR i -> rows (i, i+8) split across half-waves.
  const int col0 = n_blk + n_off + ln;
  float biasv[4];
#pragma unroll
  for (int nj = 0; nj < 4; ++nj) biasv[nj] = bias ? bias[col0 + 16 * nj] : 0.0f;
#pragma unroll
  for (int mi = 0; mi < 2; ++mi) {
#pragma unroll
    for (int i = 0; i < 8; ++i) {
      const int r = m_blk + m_off + 16 * mi + i + 8 * half;
#pragma unroll
      for (int nj = 0; nj < 4; ++nj) {
        const int cc = col0 + 16 * nj;
        float v = acc[mi][nj][i] + biasv[nj];
        if constexpr (EPI == EPI_GELU || EPI == EPI_GELU_RES) v = gelu_exact(v);
        if constexpr (EPI == EPI_RES || EPI == EPI_GELU_RES)
          v += resb[(size_t)r * N + cc];
        C[(size_t)r * N + cc] = v;
      }
    }
  }
}

// --------------- AFT elementwise stages ------------------------------------
__global__ void colmax_kernel(const float* __restrict__ K, float* __restrict__ maxK) {
  const int i = blockIdx.x * blockDim.x + threadIdx.x;  // over T*D
  float m = -INFINITY;
#pragma unroll
  for (int b = 0; b < Bc; ++b) m = fmaxf(m, K[(size_t)b * Tc * Dc + i]);
  maxK[i] = m;
}

__global__ void expkv_kernel(const float* __restrict__ K, const float* __restrict__ V,
                             const float* __restrict__ maxK, float* __restrict__ Bm) {
  const size_t i = (size_t)blockIdx.x * blockDim.x + threadIdx.x;  // over B*T*D
  const int d = (int)(i % Dc);
  const size_t bt = i / Dc;
  const int t = (int)(bt % Tc);
  const float ek = expf(K[i] - maxK[(size_t)t * Dc + d]);
  const size_t o = bt * (size_t)(2 * Dc);
  Bm[o + d]      = ek * V[i];
  Bm[o + Dc + d] = ek;
}

__global__ void expw_kernel(const float* __restrict__ w, float* __restrict__ ew) {
  const int row = blockIdx.x;
  const float* wr = w + (size_t)row * Tc;
  float m = -INFINITY;
  for (int i = threadIdx.x; i < Tc; i += blockDim.x) m = fmaxf(m, wr[i]);
  m = wave_max(m);
  __shared__ float sm[8];
  if ((threadIdx.x & 31) == 0) sm[threadIdx.x >> 5] = m;
  __syncthreads();
  float mm = sm[0];
#pragma unroll
  for (int j = 1; j < 8; ++j) mm = fmaxf(mm, sm[j]);
  float* er = ew + (size_t)row * Tc;
  for (int i = threadIdx.x; i < Tc; i += blockDim.x) er[i] = expf(wr[i] - mm);
}

__global__ void gate_kernel(const float* __restrict__ ND, const float* __restrict__ Q,
                            float* __restrict__ attn) {
  const size_t i = (size_t)blockIdx.x * blockDim.x + threadIdx.x;  // over B*T*D
  const int d = (int)(i % Dc);
  const size_t bt = i / Dc;
  const float num = ND[bt * (size_t)(2 * Dc) + d];
  const float den = ND[bt * (size_t)(2 * Dc) + Dc + d];
  const float q = Q[i];
  attn[i] = (1.0f / (1.0f + expf(-q))) * (num / den);
}

}  // namespace

// ---------------------------------------------------------------------------
extern "C" void kernel_launch(void* const* d_in, const int* in_sizes, int n_in,
                              void* d_out, int out_size, void* d_ws, size_t ws_size,
                              hipStream_t stream) {
  (void)in_sizes; (void)n_in; (void)out_size; (void)ws_size;
  const float* x     = (const float*)d_in[0];
  const float* ln1_g = (const float*)d_in[1];
  const float* ln1_b = (const float*)d_in[2];
  const float* Wk    = (const float*)d_in[3];
  const float* bk    = (const float*)d_in[4];
  const float* Wv    = (const float*)d_in[5];
  const float* bv    = (const float*)d_in[6];
  const float* Wq    = (const float*)d_in[7];
  const float* bq    = (const float*)d_in[8];
  const float* w     = (const float*)d_in[9];
  const float* Wo    = (const float*)d_in[10];
  const float* bo    = (const float*)d_in[11];
  const float* ln2_g = (const float*)d_in[12];
  const float* ln2_b = (const float*)d_in[13];
  const float* W1    = (const float*)d_in[14];
  const float* b1    = (const float*)d_in[15];
  const float* W2    = (const float*)d_in[16];
  const float* b2    = (const float*)d_in[17];
  float* out = (float*)d_out;

  const size_t NT = (size_t)Bc * Tc;  // 16384 rows
  const size_t MD = NT * Dc;          // 8M floats
  float* ws   = (float*)d_ws;
  float* h    = ws;                         // [NT, D]   LN1 out (reused as h2)
  float* Kb   = h    + MD;                  // [NT, D]   K (reused as attn_in)
  float* Vb   = Kb   + MD;                  // [NT, D]   V (reused as out1)
  float* Qb   = Vb   + MD;                  // [NT, D]   Q
  float* maxK = Qb   + MD;                  // [T, D]
  float* expw = maxK + (size_t)Tc * Dc;     // [T, T]
  float* Bm   = expw + (size_t)Tc * Tc;     // [B, T, 2D] packed (eK*V | eK)
  float* Bmt  = Bm   + NT * (size_t)(2*Dc); // [B, 2D, T] transposed
  float* WkT  = Bmt  + NT * (size_t)(2*Dc); // [D, D]
  float* WvT  = WkT  + (size_t)Dc * Dc;
  float* WqT  = WvT  + (size_t)Dc * Dc;
  float* WoT  = WqT  + (size_t)Dc * Dc;
  float* W1T  = WoT  + (size_t)Dc * Dc;     // [H, D]
  float* W2T  = W1T  + (size_t)Dc * Hc;     // [D, H]
  float* ND   = Bm;                         // reuse: einsum out (num | den)
  float* attn = Kb;                         // reuse
  float* out1 = Vb;                         // reuse
  float* h2   = h;                          // reuse
  float* hH   = Bm;                         // reuse Bm+Bmt (32M floats): [NT, H]

  const dim3 blk(256);
  const long long bTD2 = (long long)Tc * 2 * Dc;

  // 0) Pre-transpose all GEMM right operands to [N][K].
  transpose_kernel<<<dim3(16, 16, 1), blk, 0, stream>>>(Wk, WkT, Dc, Dc, 0, 0);
  transpose_kernel<<<dim3(16, 16, 1), blk, 0, stream>>>(Wv, WvT, Dc, Dc, 0, 0);
  transpose_kernel<<<dim3(16, 16, 1), blk, 0, stream>>>(Wq, WqT, Dc, Dc, 0, 0);
  transpose_kernel<<<dim3(16, 16, 1), blk, 0, stream>>>(Wo, WoT, Dc, Dc, 0, 0);
  transpose_kernel<<<dim3(Hc / 32, 16, 1), blk, 0, stream>>>(W1, W1T, Dc, Hc, 0, 0);
  transpose_kernel<<<dim3(16, Hc / 32, 1), blk, 0, stream>>>(W2, W2T, Hc, Dc, 0, 0);

  // 1) h = LN1(x)
  ln_kernel<<<dim3(NT / 8), blk, 0, stream>>>(x, ln1_g, ln1_b, h);

  // 2) K, V, Q projections: [NT,512] x [512,512]
  const dim3 g_proj(Dc / 128, NT / 128, 1);
  gemm_wmma_f32<EPI_NONE><<<g_proj, blk, 0, stream>>>(h, WkT, bk, nullptr, Kb,
                                                      (int)NT, Dc, Dc, 0, 0, 0);
  gemm_wmma_f32<EPI_NONE><<<g_proj, blk, 0, stream>>>(h, WvT, bv, nullptr, Vb,
                                                      (int)NT, Dc, Dc, 0, 0, 0);
  gemm_wmma_f32<EPI_NONE><<<g_proj, blk, 0, stream>>>(h, WqT, bq, nullptr, Qb,
                                                      (int)NT, Dc, Dc, 0, 0, 0);

  // 3) AFT stabilization + packed right operand (then transpose per batch)
  colmax_kernel<<<dim3((Tc * Dc) / 256), blk, 0, stream>>>(Kb, maxK);
  expkv_kernel<<<dim3((unsigned)(NT * Dc / 256)), blk, 0, stream>>>(Kb, Vb, maxK, Bm);
  transpose_kernel<<<dim3((2 * Dc) / 32, Tc / 32, Bc), blk, 0, stream>>>(
      Bm, Bmt, Tc, 2 * Dc, bTD2, bTD2);
  expw_kernel<<<dim3(Tc), blk, 0, stream>>>(w, expw);

  // 4) Fused num|den einsum per batch: expw [T,T] x Bmt^T -> ND [T,2D]
  const dim3 g_aft((2 * Dc) / 128, Tc / 128, Bc);
  gemm_wmma_f32<EPI_NONE><<<g_aft, blk, 0, stream>>>(expw, Bmt, nullptr, nullptr, ND,
                                                     Tc, 2 * Dc, Tc, 0, bTD2, bTD2);

  // 5) gate: attn = sigmoid(Q) * num/den
  gate_kernel<<<dim3((unsigned)(NT * Dc / 256)), blk, 0, stream>>>(ND, Qb, attn);

  // 6) out1 = attn @ Wo + bo + x
  gemm_wmma_f32<EPI_RES><<<g_proj, blk, 0, stream>>>(attn, WoT, bo, x, out1,
                                                     (int)NT, Dc, Dc, 0, 0, 0);

  // 7) h2 = LN2(out1)
  ln_kernel<<<dim3(NT / 8), blk, 0, stream>>>(out1, ln2_g, ln2_b, h2);

  // 8) hH = gelu(h2 @ W1 + b1): [NT,512] x [512,2048]
  const dim3 g_ffn1(Hc / 128, NT / 128, 1);
  gemm_wmma_f32<EPI_GELU><<<g_ffn1, blk, 0, stream>>>(h2, W1T, b1, nullptr, hH,
                                                      (int)NT, Hc, Dc, 0, 0, 0);

  // 9) out = gelu(hH @ W2 + b2) + out1: [NT,2048] x [2048,512]
  gemm_wmma_f32<EPI_GELU_RES><<<g_proj, blk, 0, stream>>>(hH, W2T, b2, out1, out,
                                                          (int)NT, Dc, Hc, 0, 0, 0);
}